// GAT_16518444220920
// MI455X (gfx1250) — compile-verified
//
#include <hip/hip_runtime.h>

#define DD 64
#define LRELU_ALPHA 0.2f

typedef __attribute__((ext_vector_type(2))) float v2f;
typedef __attribute__((ext_vector_type(8))) float v8f;

// ---- order-preserving float <-> uint key for atomicMax-based segment max ----
__device__ __forceinline__ unsigned fkey(float f) {
    unsigned u = __float_as_uint(f);
    return (u & 0x80000000u) ? ~u : (u | 0x80000000u);
}
__device__ __forceinline__ float funkey(unsigned k) {
    unsigned u = (k & 0x80000000u) ? (k & 0x7FFFFFFFu) : ~k;
    return __uint_as_float(u);
}

// ---------------------------------------------------------------------------
// h = X @ W   (X: [N,64] row-major, W: [64,64] row-major, h: [N,64])
// One wave32 per 16x16 output tile, K-loop of 16 steps of V_WMMA_F32_16X16X4_F32.
// A (16x4 f32) layout: lane = m + 16*khalf; v0 = K=(0|2), v1 = K=(1|3).
// B (4x16 f32) layout mirrored: lane = n + 16*khalf; v0/v1 = rows K.
// C/D (16x16 f32): VGPR r, lane L -> row r + 8*(L/16), col L%16.
// ---------------------------------------------------------------------------
__global__ void gat_gemm_wmma_f32(const float* __restrict__ X,
                                  const float* __restrict__ W,
                                  float* __restrict__ H,
                                  int N, int Mtiles)
{
    int wid  = (blockIdx.x * blockDim.x + threadIdx.x) >> 5;  // wave-uniform
    int lane = threadIdx.x & 31;
    if (wid >= Mtiles * 4) return;            // uniform exit: EXEC stays all-1s

    int mt   = wid >> 2;
    int nt   = wid & 3;
    int half = lane >> 4;                     // which K-half this lane holds
    int r    = lane & 15;

    int mrow   = mt * 16 + r;
    int mclamp = mrow < N ? mrow : N - 1;     // clamp loads, predicate stores
    int ncol   = nt * 16 + r;

    v8f c = {0.f, 0.f, 0.f, 0.f, 0.f, 0.f, 0.f, 0.f};

#pragma unroll
    for (int ks = 0; ks < 16; ++ks) {
        int k = ks * 4 + 2 * half;
        v2f a, b;
        a.x = X[mclamp * DD + k];
        a.y = X[mclamp * DD + k + 1];
        b.x = W[k * DD + ncol];
        b.y = W[(k + 1) * DD + ncol];
        c = __builtin_amdgcn_wmma_f32_16x16x4_f32(false, a, false, b,
                                                  (short)0, c, false, false);
    }

#pragma unroll
    for (int vr = 0; vr < 8; ++vr) {
        int row = mt * 16 + vr + 8 * half;
        if (row < N) H[row * DD + ncol] = c[vr];
    }
}

// s_src[n] = h[n,:] . a_s ; s_dst[n] = h[n,:] . a_d
__global__ void gat_scores(const float* __restrict__ H,
                           const float* __restrict__ a_s,
                           const float* __restrict__ a_d,
                           float* __restrict__ Ss, float* __restrict__ Sd, int N)
{
    int n = blockIdx.x * blockDim.x + threadIdx.x;
    if (n >= N) return;
    float ss = 0.f, sd = 0.f;
#pragma unroll 8
    for (int d = 0; d < DD; ++d) {
        float hv = H[n * DD + d];
        ss += hv * a_s[d];      // a_s/a_d uniform -> scalar loads
        sd += hv * a_d[d];
    }
    Ss[n] = ss;
    Sd[n] = sd;
}

// pass 1: e = leaky_relu(s_src[src]+s_dst[dst]); segment max over dst via keyed atomicMax
__global__ void gat_edge_max(const int* __restrict__ src, const int* __restrict__ dst,
                             const float* __restrict__ Ss, const float* __restrict__ Sd,
                             float* __restrict__ Ex, unsigned* __restrict__ Mu, int E)
{
    int e = blockIdx.x * blockDim.x + threadIdx.x;
    if (e >= E) return;
    int s = src[e], d = dst[e];
    float v = Ss[s] + Sd[d];
    v = v > 0.f ? v : LRELU_ALPHA * v;
    Ex[e] = v;
    atomicMax(&Mu[d], fkey(v));
}

// pass 2: ex = exp(e - m[dst]); z[dst] += ex
__global__ void gat_edge_exp(const int* __restrict__ dst,
                             float* __restrict__ Ex,
                             const unsigned* __restrict__ Mu,
                             float* __restrict__ Z, int E)
{
    int e = blockIdx.x * blockDim.x + threadIdx.x;
    if (e >= E) return;
    int d = dst[e];
    float m = funkey(Mu[d]);
    float x = __expf(Ex[e] - m);
    Ex[e] = x;
    atomicAdd(&Z[d], x);
}

// pass 3: acc[dst,:] += (ex/z[dst]) * h[src,:] ; one wave32 per edge, float2/lane
__global__ void gat_edge_scatter(const int* __restrict__ src, const int* __restrict__ dst,
                                 const float* __restrict__ H,
                                 const float* __restrict__ Ex, const float* __restrict__ Z,
                                 float* __restrict__ Acc, int E)
{
    int gw   = (blockIdx.x * blockDim.x + threadIdx.x) >> 5;  // edge id (wave-uniform)
    int lane = threadIdx.x & 31;
    if (gw >= E) return;
    int   s    = src[gw];
    int   d    = dst[gw];
    float attn = Ex[gw] / (Z[d] + 1e-16f);
    const float2* hp = (const float2*)(H + (size_t)s * DD);
    float2 hv = hp[lane];                                     // 256B coalesced, L2-resident
    atomicAdd(&Acc[d * DD + 2 * lane],     hv.x * attn);
    atomicAdd(&Acc[d * DD + 2 * lane + 1], hv.y * attn);
}

__global__ void gat_elu(const float* __restrict__ in, float* __restrict__ out, int n)
{
    int i = blockIdx.x * blockDim.x + threadIdx.x;
    if (i >= n) return;
    float v = in[i];
    out[i] = v > 0.f ? v : __expf(v) - 1.f;
}

// ---------------------------------------------------------------------------

struct GatWS {
    float* b0; float* b1; float* Hbuf;
    float* Ss; float* Sd; float* Z; unsigned* Mu; float* Ex;
    const int* src; const int* dst;
    int N; int E;
};

static void run_conv(const GatWS& w, const float* xin,
                     const float* Wp, const float* asp, const float* adp,
                     float* acc, hipStream_t stream)
{
    int N = w.N, E = w.E;
    hipMemsetAsync(acc,  0, (size_t)N * DD * sizeof(float), stream);
    hipMemsetAsync(w.Mu, 0, (size_t)N * sizeof(unsigned),   stream);
    hipMemsetAsync(w.Z,  0, (size_t)N * sizeof(float),      stream);

    int Mtiles  = (N + 15) / 16;
    int threads = Mtiles * 4 * 32;
    gat_gemm_wmma_f32<<<(threads + 255) / 256, 256, 0, stream>>>(xin, Wp, w.Hbuf, N, Mtiles);
    gat_scores<<<(N + 255) / 256, 256, 0, stream>>>(w.Hbuf, asp, adp, w.Ss, w.Sd, N);
    gat_edge_max<<<(E + 255) / 256, 256, 0, stream>>>(w.src, w.dst, w.Ss, w.Sd, w.Ex, w.Mu, E);
    gat_edge_exp<<<(E + 255) / 256, 256, 0, stream>>>(w.dst, w.Ex, w.Mu, w.Z, E);
    long long sthreads = (long long)E * 32;
    gat_edge_scatter<<<(int)((sthreads + 255) / 256), 256, 0, stream>>>(
        w.src, w.dst, w.Hbuf, w.Ex, w.Z, acc, E);
}

extern "C" void kernel_launch(void* const* d_in, const int* in_sizes, int n_in,
                              void* d_out, int out_size, void* d_ws, size_t ws_size,
                              hipStream_t stream)
{
    const float* feat = (const float*)d_in[0];
    const int*   EI   = (const int*)  d_in[1];
    const float* W1   = (const float*)d_in[2];
    const float* As1  = (const float*)d_in[3];
    const float* Ad1  = (const float*)d_in[4];
    const float* W2   = (const float*)d_in[5];
    const float* As2  = (const float*)d_in[6];
    const float* Ad2  = (const float*)d_in[7];
    float* out = (float*)d_out;

    int N = in_sizes[0] / DD;        // 50000
    int E = in_sizes[1] / 2;         // 800000

    // workspace carve-up (all 256B aligned)
    char*  ws = (char*)d_ws;
    size_t nd = (size_t)N * DD * sizeof(float);
    size_t o  = 0;
    auto alignup = [](size_t x) { return (x + 255) & ~(size_t)255; };

    GatWS w;
    w.b0   = (float*)(ws + o);    o += alignup(nd);
    w.b1   = (float*)(ws + o);    o += alignup(nd);
    w.Hbuf = (float*)(ws + o);    o += alignup(nd);
    w.Ss   = (float*)(ws + o);    o += alignup((size_t)N * 4);
    w.Sd   = (float*)(ws + o);    o += alignup((size_t)N * 4);
    w.Z    = (float*)(ws + o);    o += alignup((size_t)N * 4);
    w.Mu   = (unsigned*)(ws + o); o += alignup((size_t)N * 4);
    w.Ex   = (float*)(ws + o);    o += alignup((size_t)E * 4);
    w.src  = EI;
    w.dst  = EI + E;
    w.N = N; w.E = E;

    const size_t WS = (size_t)DD * DD;   // per-head W stride
    int nElems = N * DD;

    // ---- layer 1: 4 sequential convs, then ELU ----
    run_conv(w, feat,  W1 + 0 * WS, As1 + 0 * DD, Ad1 + 0 * DD, w.b0, stream);
    run_conv(w, w.b0,  W1 + 1 * WS, As1 + 1 * DD, Ad1 + 1 * DD, w.b1, stream);
    run_conv(w, w.b1,  W1 + 2 * WS, As1 + 2 * DD, Ad1 + 2 * DD, w.b0, stream);
    run_conv(w, w.b0,  W1 + 3 * WS, As1 + 3 * DD, Ad1 + 3 * DD, w.b1, stream);
    gat_elu<<<(nElems + 255) / 256, 256, 0, stream>>>(w.b1, w.b1, nElems);

    // ---- layer 2: 4 sequential convs, then ELU into d_out ----
    run_conv(w, w.b1,  W2 + 0 * WS, As2 + 0 * DD, Ad2 + 0 * DD, w.b0, stream);
    run_conv(w, w.b0,  W2 + 1 * WS, As2 + 1 * DD, Ad2 + 1 * DD, w.b1, stream);
    run_conv(w, w.b1,  W2 + 2 * WS, As2 + 2 * DD, Ad2 + 2 * DD, w.b0, stream);
    run_conv(w, w.b0,  W2 + 3 * WS, As2 + 3 * DD, Ad2 + 3 * DD, w.b1, stream);
    gat_elu<<<(nElems + 255) / 256, 256, 0, stream>>>(w.b1, out, nElems);
}